// LinearRNN_1408749273481
// MI455X (gfx1250) — compile-verified
//
#include <hip/hip_runtime.h>

// ============================================================================
// LinearRNN: ys[t] = A^(t+1) @ y0,  D=2048, STEPS=8192, fp32.
//
// MI455X strategy:
//   - fp32 WMMA (V_WMMA_F32_16X16X4_F32): 8192-step orthogonal recurrence
//     needs fp32; error ~ sqrt(D)*eps*sqrt(steps) ~ 3e-4.
//   - Doubling ladder: after squaring W_n -> W_2n, extend output rows
//     rows[n..2n) = rows[0..n) @ W_n. Only 29 dependent launches total,
//     and the 7 tail GEMMs are M=1024 (256 blocks -> good WGP occupancy).
//   - Pair-interleaved LDS panels so every WMMA fragment is one aligned
//     ds_load_b64 into an even VGPR pair (no v_mov shuffling); row pads
//     ≡ 32 (mod 64) dwords make the two half-wave K-offsets bank-disjoint.
//   - Register double-buffering of global->LDS panel loads.
// ============================================================================

typedef __attribute__((ext_vector_type(2))) float v2f;
typedef __attribute__((ext_vector_type(8))) float v8f;

#define DIM   2048
#define NSTEP 8192

// ---------------------------------------------------------------------------
// 32x32 tiled transpose: T = A^T
// ---------------------------------------------------------------------------
__global__ void xpose_k(const float* __restrict__ A, float* __restrict__ T) {
  __shared__ float tile[32][33];
  const int bx = blockIdx.x * 32, by = blockIdx.y * 32;
  #pragma unroll
  for (int i = 0; i < 4; ++i)
    tile[threadIdx.y + i * 8][threadIdx.x] =
        A[(size_t)(by + threadIdx.y + i * 8) * DIM + bx + threadIdx.x];
  __syncthreads();
  #pragma unroll
  for (int i = 0; i < 4; ++i)
    T[(size_t)(bx + threadIdx.y + i * 8) * DIM + by + threadIdx.x] =
        tile[threadIdx.x][threadIdx.y + i * 8];
}

// ---------------------------------------------------------------------------
// y = A @ x : one wave32 per row, 8 rows per 256-thread block.
// ---------------------------------------------------------------------------
__global__ void matvec_k(const float* __restrict__ A, const float* __restrict__ x,
                         float* __restrict__ y) {
  const int row  = blockIdx.x * 8 + (threadIdx.x >> 5);
  const int lane = threadIdx.x & 31;
  const float* ar = A + (size_t)row * DIM;
  float s = 0.f;
  #pragma unroll
  for (int j = 0; j < DIM / 128; ++j) {
    const float4 a = *(const float4*)(ar + j * 128 + lane * 4);
    const float4 b = *(const float4*)(x + j * 128 + lane * 4);
    s += a.x * b.x + a.y * b.y + a.z * b.z + a.w * b.w;
  }
  #pragma unroll
  for (int off = 16; off > 0; off >>= 1) s += __shfl_xor(s, off, 32);
  if (lane == 0) y[row] = s;
}

// ---------------------------------------------------------------------------
// C[M x 2048] = X[M x 2048] @ W[2048 x 2048], row-major fp32, ld = 2048.
// Block: 64(M) x 128(N), 4 waves (2x2), wave tile 32x64 -> 8 v8f accumulators.
// LDS layout (pair-interleaved, j = K-pair index within the 32-wide panel):
//   As[j][2m+p] = X[mBase+m][kb+2j+p]   (row stride 160 ≡ 32 mod 64 dwords)
//   Bs[j][2n+p] = W[kb+2j+p][nBase+n]   (row stride 288 ≡ 32 mod 64 dwords)
// Fragment for half h (lanes 0-15: h=0 -> K+0/1; lanes 16-31: h=1 -> K+2/3)
// is a single conflict-free ds_load_b64 at row j = kstep*2 + h.
// ---------------------------------------------------------------------------
__global__ __launch_bounds__(128)
void gemm_f32_k(const float* __restrict__ X, const float* __restrict__ W,
                float* __restrict__ C) {
  __shared__ float As[16][160];
  __shared__ float Bs[16][288];

  const int tid  = threadIdx.x;
  const int lane = tid & 31;
  const int wave = tid >> 5;
  const int wm   = wave & 1;      // wave row  (0..1)
  const int wn   = wave >> 1;     // wave col  (0..1)
  const int half = lane >> 4;
  const int l16  = lane & 15;
  const int mBase = blockIdx.y * 64;
  const int nBase = blockIdx.x * 128;

  // A-panel loader: f = tid + 128*i (i=0..7): m = f&63, j = f>>6
  const int am  = tid & 63;
  const int aj0 = tid >> 6;          // 0..1 ; j = aj0 + 2*i
  // B-panel loader: f = tid + 128*i (i=0..3): j = f>>5, n4 = (f&31)*4
  const int bn4 = (tid & 31) << 2;
  const int bj0 = tid >> 5;          // 0..3 ; j = bj0 + 4*i

  float2 aReg[8];
  float4 bReg0[4], bReg1[4];

  auto loadPanel = [&](int kb) {
    #pragma unroll
    for (int i = 0; i < 8; ++i) {
      const int j = aj0 + 2 * i;
      aReg[i] = *(const float2*)(X + (size_t)(mBase + am) * DIM + kb + 2 * j);
    }
    #pragma unroll
    for (int i = 0; i < 4; ++i) {
      const int j = bj0 + 4 * i;
      bReg0[i] = *(const float4*)(W + (size_t)(kb + 2 * j) * DIM + nBase + bn4);
      bReg1[i] = *(const float4*)(W + (size_t)(kb + 2 * j + 1) * DIM + nBase + bn4);
    }
  };
  auto storePanel = [&]() {
    #pragma unroll
    for (int i = 0; i < 8; ++i)                     // conflict-free b64 stores
      *(float2*)&As[aj0 + 2 * i][2 * am] = aReg[i];
    #pragma unroll
    for (int i = 0; i < 4; ++i) {
      const int j = bj0 + 4 * i;
      #pragma unroll
      for (int t = 0; t < 4; ++t) {
        float2 v;
        v.x = ((const float*)&bReg0[i])[t];
        v.y = ((const float*)&bReg1[i])[t];
        *(float2*)&Bs[j][2 * (bn4 + t)] = v;
      }
    }
  };

  v8f acc[2][4];
  #pragma unroll
  for (int mt = 0; mt < 2; ++mt)
    #pragma unroll
    for (int nt = 0; nt < 4; ++nt)
      #pragma unroll
      for (int g = 0; g < 8; ++g) acc[mt][nt][g] = 0.f;

  loadPanel(0);
  for (int kb = 0; kb < DIM; kb += 32) {
    storePanel();
    __syncthreads();
    if (kb + 32 < DIM) loadPanel(kb + 32);   // prefetch next panel into regs

    #pragma unroll
    for (int ks = 0; ks < 8; ++ks) {         // K-step of 4: pair row j
      const int j = 2 * ks + half;
      v2f a[2], b[4];
      #pragma unroll
      for (int mt = 0; mt < 2; ++mt)
        a[mt] = *(const v2f*)&As[j][2 * (wm * 32 + mt * 16 + l16)];
      #pragma unroll
      for (int nt = 0; nt < 4; ++nt)
        b[nt] = *(const v2f*)&Bs[j][2 * (wn * 64 + nt * 16 + l16)];
      #pragma unroll
      for (int mt = 0; mt < 2; ++mt)
        #pragma unroll
        for (int nt = 0; nt < 4; ++nt)
          acc[mt][nt] = __builtin_amdgcn_wmma_f32_16x16x4_f32(
              false, a[mt], false, b[nt], (short)0, acc[mt][nt], false, false);
    }
    __syncthreads();
  }

  // Epilogue: C/D layout -> row = 8*half + g, col = l16 within each 16x16 tile
  #pragma unroll
  for (int mt = 0; mt < 2; ++mt)
    #pragma unroll
    for (int nt = 0; nt < 4; ++nt) {
      const int col = nBase + wn * 64 + nt * 16 + l16;
      #pragma unroll
      for (int g = 0; g < 8; ++g) {
        const int row = mBase + wm * 32 + mt * 16 + half * 8 + g;
        C[(size_t)row * DIM + col] = acc[mt][nt][g];
      }
    }
}

// ---------------------------------------------------------------------------
extern "C" void kernel_launch(void* const* d_in, const int* in_sizes, int n_in,
                              void* d_out, int out_size, void* d_ws, size_t ws_size,
                              hipStream_t stream) {
  (void)in_sizes; (void)n_in; (void)out_size;
  const float* A      = (const float*)d_in[0];   // [2048, 2048]
  const float* y_init = (const float*)d_in[1];   // [2, 2048]; row 0 is the seed
  float* out = (float*)d_out;                    // [8192, 2048]

  const size_t MAT = (size_t)DIM * DIM;          // 16 MB
  float* bufA = (float*)d_ws;                    // persistent (final W here)
  float* bufB = (ws_size >= 2 * MAT * sizeof(float))
                    ? bufA + MAT
                    : out + (size_t)(NSTEP - DIM) * DIM;  // tail: dead before
                                                          // phase-2 reaches it
  // ---- row 0 and W1 = A^T ---------------------------------------------
  matvec_k<<<DIM / 8, 256, 0, stream>>>(A, y_init, out);
  xpose_k<<<dim3(64, 64), dim3(32, 8), 0, stream>>>(A, bufA);

  // ---- doubling ladder: n = 1,2,4,...,512 ------------------------------
  // 1) rows[n..2n) = rows[0..n) @ W_n  (row-wise independent; rows beyond
  //    the valid prefix are garbage and are overwritten by later launches,
  //    so the final d_out state is deterministic).
  // 2) W_2n = W_n @ W_n  ((A^2n)^T = (A^n)^T (A^n)^T).
  float* Wcur = bufA;
  float* Wnxt = bufB;
  int n = 1;
  for (int i = 0; i < 10; ++i) {
    const int M = (n < 64) ? 64 : n;
    gemm_f32_k<<<dim3(DIM / 128, M / 64), 128, 0, stream>>>(
        out, Wcur, out + (size_t)n * DIM);
    gemm_f32_k<<<dim3(DIM / 128, DIM / 64), 128, 0, stream>>>(Wcur, Wcur, Wnxt);
    float* t = Wcur; Wcur = Wnxt; Wnxt = t;
    n *= 2;
  }
  // n == 1024; 10 swaps (even) -> Wcur = (A^1024)^T sits in bufA (d_ws).

  // ---- tail: 7 dependent GEMMs, M = 1024 (256 blocks each) -------------
  for (int j = 0; j + 1 < NSTEP / 1024; ++j)
    gemm_f32_k<<<dim3(DIM / 128, 1024 / 64), 128, 0, stream>>>(
        out + (size_t)j * 1024 * DIM, Wcur, out + (size_t)(j + 1) * 1024 * DIM);
}